// DeepFit_26663156973927
// MI455X (gfx1250) — compile-verified
//
#include <hip/hip_runtime.h>
#include <math.h>

#define NPTS 1024
#define KNB1 40
#define KNB2 20

typedef float v2f __attribute__((ext_vector_type(2)));
typedef float v8f __attribute__((ext_vector_type(8)));

__device__ __forceinline__ v8f wmma4(v2f a, v2f b, v8f c) {
    // D = A(16x4 f32) * B(4x16 f32) + C(16x16 f32)
    return __builtin_amdgcn_wmma_f32_16x16x4_f32(false, a, false, b, (short)0, c,
                                                 false, false);
}

__device__ __forceinline__ float leaky(float x) { return x >= 0.f ? x : 0.2f * x; }

// Packed A-operand index: element (row m, col i) of a 64x64 weight ->
// float2 slot (kk*2+hi)*64+m, component i&1   (k = kk*4 + 2*hi + r)
__device__ __forceinline__ int pkidx(int m, int i) {
    return ((((i >> 2) * 2 + ((i >> 1) & 1)) * 64 + m) << 1) | (i & 1);
}

// ---------------------------------------------------------------- prep params
struct PrepArgs {
    const float *g[6], *b[6], *m[6], *v[6];
    const float *W1, *W2, *W3, *W4, *W5;
    float *sb;                               // [6][128]: scale | bias
    float *W1ap, *W1dp, *W3ap, *W3dp, *W5bp; // packed A for gemm64
    float *W2p, *W4p, *W5ap;                 // packed A for fused kernels
};

__global__ void k_prep(PrepArgs pa) {
    int o = threadIdx.x;  // 64 threads
    for (int l = 0; l < 6; ++l) {
        float s = pa.g[l][o] / sqrtf(pa.v[l][o] + 1e-5f);
        pa.sb[l * 128 + o]      = s;
        pa.sb[l * 128 + 64 + o] = pa.b[l][o] - pa.m[l][o] * s;
    }
    for (int i = 0; i < 64; ++i) {
        int q = pkidx(o, i);
        float a1 = pa.W1[o * 128 + i], b1 = pa.W1[o * 128 + 64 + i];
        pa.W1ap[q] = a1;  pa.W1dp[q] = b1 - a1;
        float a3 = pa.W3[o * 128 + i], b3 = pa.W3[o * 128 + 64 + i];
        pa.W3ap[q] = a3;  pa.W3dp[q] = b3 - a3;
        pa.W5ap[q] = pa.W5[o * 128 + i];
        pa.W5bp[q] = pa.W5[o * 128 + 64 + i];
        pa.W2p[q]  = pa.W2[o * 64 + i];
        pa.W4p[q]  = pa.W4[o * 64 + i];
    }
}

// ------------------------------------------- conv0 + bn0 (writes packed x0)
// x0p layout: float2 slot t*1024+n holds {x0[2t][n], x0[2t+1][n]}
__global__ __launch_bounds__(256) void k_conv0(const float* __restrict__ pts,
                                               const float* __restrict__ W0,
                                               const float* __restrict__ sb,
                                               float* __restrict__ x0p) {
    int gid = blockIdx.x * 256 + threadIdx.x;      // B*64*1024
    int b = gid >> 16, r = gid & 65535;
    int o = r >> 10, n = r & 1023;
    const float* pb = pts + b * 3 * NPTS;
    float acc = W0[o * 3 + 0] * pb[n] + W0[o * 3 + 1] * pb[NPTS + n] +
                W0[o * 3 + 2] * pb[2 * NPTS + n];
    x0p[b * 65536 + (o >> 1) * 2048 + n * 2 + (o & 1)] = leaky(sb[o] * acc + sb[64 + o]);
}

// ---------------------------------------------------------------- ||x||^2
__global__ __launch_bounds__(256) void k_xx(const float* __restrict__ x0p,
                                            float* __restrict__ xx) {
    int t = blockIdx.x * 256 + threadIdx.x;        // B*N
    int b = t >> 10, n = t & 1023;
    const v2f* xb = (const v2f*)(x0p + b * 65536);
    float s = 0.f;
    for (int c = 0; c < 32; ++c) { v2f v = xb[c * 1024 + n]; s += v.x * v.x + v.y * v.y; }
    xx[t] = s;
}

// --------------------------------------- Gram (WMMA) + fused top-40 selection
__global__ __launch_bounds__(256) void k_gram_topk(const float* __restrict__ x0p,
                                                   const float* __restrict__ xx,
                                                   int* __restrict__ idxout) {
    extern __shared__ float pd[];                  // 16 rows x 1024 cols (64 KB)
    int b = blockIdx.y;
    int i0 = blockIdx.x * 16;
    int tid = threadIdx.x;
    int w = tid >> 5, lane = tid & 31, lm = lane & 15, hi = lane >> 4;
    const v2f* xp = (const v2f*)(x0p + b * 65536);
    const float* xxb = xx + b * 1024;

    // Phase 1: 16x1024 tile of pd = 2*inner - xx_i - xx_j via fp32 WMMA
    for (int p = 0; p < 8; ++p) {
        int n0 = (w * 8 + p) * 16;
        v8f acc = {0.f, 0.f, 0.f, 0.f, 0.f, 0.f, 0.f, 0.f};
        for (int kk = 0; kk < 16; ++kk) {
            int t = kk * 2 + hi;
            v2f a  = xp[t * 1024 + i0 + lm];
            v2f bv = xp[t * 1024 + n0 + lm];
            acc = wmma4(a, bv, acc);
        }
        int j = n0 + lm;
        float xj = xxb[j];
        for (int r = 0; r < 8; ++r) {
            int ri = r + hi * 8;
            pd[ri * 1024 + j] = 2.f * acc[r] - xxb[i0 + ri] - xj;
        }
    }
    __syncthreads();

    // Phase 2: each wave selects top-40 for 2 rows (iterative argmax)
    volatile float* vpd = pd;
    for (int rr = 0; rr < 2; ++rr) {
        int row = w * 2 + rr;
        int gi = i0 + row;
        for (int sel = 0; sel < KNB1; ++sel) {
            float bestv = -3.4e38f; int besti = 0;
            for (int j = lane; j < 1024; j += 32) {
                float v = vpd[row * 1024 + j];
                if (v > bestv) { bestv = v; besti = j; }
            }
            for (int off = 16; off; off >>= 1) {
                float ov = __shfl_xor(bestv, off, 32);
                int   oi = __shfl_xor(besti, off, 32);
                if (ov > bestv || (ov == bestv && oi < besti)) { bestv = ov; besti = oi; }
            }
            if (lane == 0) {
                idxout[(b * 1024 + gi) * KNB1 + sel] = besti;
                vpd[row * 1024 + besti] = -3.4e38f;
            }
        }
    }
}

// ---------- OUT(64xN, row-major) = W(64x64, packed) @ X(64xN, packed), WMMA
__global__ __launch_bounds__(256) void k_gemm64(const float* __restrict__ Wpk,
                                                const float* __restrict__ Xp,
                                                float* __restrict__ OUT) {
    int tid = threadIdx.x;
    int w = tid >> 5, lane = tid & 31, lm = lane & 15, hi = lane >> 4;
    int tile = blockIdx.x * 8 + w;                 // 256 tiles per batch
    int b = tile >> 8, r = tile & 255;
    int ot = r >> 6, nt = r & 63;
    const v2f* Xb = (const v2f*)(Xp + b * 65536);
    const v2f* Wp = (const v2f*)Wpk;
    int n0 = nt * 16;
    v8f acc = {0.f, 0.f, 0.f, 0.f, 0.f, 0.f, 0.f, 0.f};
    for (int kk = 0; kk < 16; ++kk) {
        int t = kk * 2 + hi;
        v2f a  = Wp[t * 64 + ot * 16 + lm];
        v2f bv = Xb[t * 1024 + n0 + lm];
        acc = wmma4(a, bv, acc);
    }
    float* Ob = OUT + b * 65536;
    for (int rr = 0; rr < 8; ++rr) {
        int o = ot * 16 + rr + hi * 8;
        Ob[o * 1024 + n0 + lm] = acc[rr];
    }
}

// ------------- fused branch 1: gather+bn1 -> WMMA(W2)+bn2 -> max_k  => xk1p
__global__ __launch_bounds__(256) void k_fused1(const float* __restrict__ y1,
                                                const float* __restrict__ y2,
                                                const int* __restrict__ idx,
                                                const float* __restrict__ W2p,
                                                const float* __restrict__ sb,
                                                float* __restrict__ xk1p) {
    __shared__ float G[64 * 160];                  // packed pair-row layout
    int b = blockIdx.y, n0 = blockIdx.x * 4;
    int tid = threadIdx.x;
    const float* s1 = sb + 128;      const float* b1 = s1 + 64;
    const float* s2 = sb + 2 * 128;  const float* b2 = s2 + 64;
    const float* ya = y1 + b * 65536;
    const float* yb = y2 + b * 65536;

    for (int e = tid; e < 64 * 160; e += 256) {
        int i = e / 160, col = e % 160;
        int nl = col / KNB1, k = col % KNB1;
        int g = idx[(b * 1024 + n0 + nl) * KNB1 + k];
        float v = ya[i * 1024 + g] + yb[i * 1024 + n0 + nl];
        G[((i >> 1) * 160 + col) * 2 + (i & 1)] = leaky(s1[i] * v + b1[i]);
    }
    __syncthreads();

    int w = tid >> 5, lane = tid & 31, lm = lane & 15, hi = lane >> 4;
    int rt = w & 3, cb = w >> 2;
    const v2f* Wp = (const v2f*)W2p;
    v8f acc[5];
    for (int p = 0; p < 5; ++p) acc[p] = (v8f){0.f,0.f,0.f,0.f,0.f,0.f,0.f,0.f};
    for (int kk = 0; kk < 16; ++kk) {
        int t = kk * 2 + hi;
        v2f a = Wp[t * 64 + rt * 16 + lm];
        #pragma unroll
        for (int p = 0; p < 5; ++p) {
            int col = (cb + 2 * p) * 16 + lm;
            v2f bv = *(const v2f*)&G[(t * 160 + col) * 2];
            acc[p] = wmma4(a, bv, acc[p]);
        }
    }
    __syncthreads();
    for (int p = 0; p < 5; ++p) {
        int col = (cb + 2 * p) * 16 + lm;
        for (int rr = 0; rr < 8; ++rr) {
            int o = rt * 16 + rr + hi * 8;
            G[((o >> 1) * 160 + col) * 2 + (o & 1)] = leaky(s2[o] * acc[p][rr] + b2[o]);
        }
    }
    __syncthreads();
    int o = tid >> 2, nl = tid & 3;
    float m = -3.4e38f;
    for (int k = 0; k < KNB1; ++k)
        m = fmaxf(m, G[((o >> 1) * 160 + nl * KNB1 + k) * 2 + (o & 1)]);
    xk1p[b * 65536 + (o >> 1) * 2048 + (n0 + nl) * 2 + (o & 1)] = m;
}

// --- fused branch 2: gather+bn3 -> WMMA(W4)+bn4 -> WMMA(W5a)+z5+bn5 -> max_k
__global__ __launch_bounds__(256) void k_fused2(const float* __restrict__ y3a,
                                                const float* __restrict__ y3b,
                                                const int* __restrict__ idx,
                                                const float* __restrict__ W4p,
                                                const float* __restrict__ W5ap,
                                                const float* __restrict__ z5,
                                                const float* __restrict__ sb,
                                                float* __restrict__ feat) {
    __shared__ float G[64 * 160];                  // packed pair-row layout
    int b = blockIdx.y, n0 = blockIdx.x * 8;
    int tid = threadIdx.x;
    const float* s3 = sb + 3 * 128;  const float* b3 = s3 + 64;
    const float* s4 = sb + 4 * 128;  const float* b4 = s4 + 64;
    const float* s5 = sb + 5 * 128;  const float* b5 = s5 + 64;
    const float* ya = y3a + b * 65536;
    const float* yb = y3b + b * 65536;

    for (int e = tid; e < 64 * 160; e += 256) {
        int i = e / 160, col = e % 160;
        int nl = col / KNB2, k = col % KNB2;
        int g = idx[(b * 1024 + n0 + nl) * KNB1 + k];   // first 20 of sorted 40
        float v = ya[i * 1024 + g] + yb[i * 1024 + n0 + nl];
        G[((i >> 1) * 160 + col) * 2 + (i & 1)] = leaky(s3[i] * v + b3[i]);
    }
    __syncthreads();

    int w = tid >> 5, lane = tid & 31, lm = lane & 15, hi = lane >> 4;
    int rt = w & 3, cb = w >> 2;
    const v2f* W4v = (const v2f*)W4p;
    const v2f* W5v = (const v2f*)W5ap;
    v8f acc[5];

    // stage B: W4 @ G, bn4, leaky -> back into LDS (packed)
    for (int p = 0; p < 5; ++p) acc[p] = (v8f){0.f,0.f,0.f,0.f,0.f,0.f,0.f,0.f};
    for (int kk = 0; kk < 16; ++kk) {
        int t = kk * 2 + hi;
        v2f a = W4v[t * 64 + rt * 16 + lm];
        #pragma unroll
        for (int p = 0; p < 5; ++p) {
            int col = (cb + 2 * p) * 16 + lm;
            v2f bv = *(const v2f*)&G[(t * 160 + col) * 2];
            acc[p] = wmma4(a, bv, acc[p]);
        }
    }
    __syncthreads();
    for (int p = 0; p < 5; ++p) {
        int col = (cb + 2 * p) * 16 + lm;
        for (int rr = 0; rr < 8; ++rr) {
            int o = rt * 16 + rr + hi * 8;
            G[((o >> 1) * 160 + col) * 2 + (o & 1)] = leaky(s4[o] * acc[p][rr] + b4[o]);
        }
    }
    __syncthreads();

    // stage C: W5a @ T + z5, bn5, leaky -> LDS (packed)
    for (int p = 0; p < 5; ++p) acc[p] = (v8f){0.f,0.f,0.f,0.f,0.f,0.f,0.f,0.f};
    for (int kk = 0; kk < 16; ++kk) {
        int t = kk * 2 + hi;
        v2f a = W5v[t * 64 + rt * 16 + lm];
        #pragma unroll
        for (int p = 0; p < 5; ++p) {
            int col = (cb + 2 * p) * 16 + lm;
            v2f bv = *(const v2f*)&G[(t * 160 + col) * 2];
            acc[p] = wmma4(a, bv, acc[p]);
        }
    }
    __syncthreads();
    const float* zb = z5 + b * 65536;
    for (int p = 0; p < 5; ++p) {
        int col = (cb + 2 * p) * 16 + lm;
        int nl = col / KNB2;
        for (int rr = 0; rr < 8; ++rr) {
            int o = rt * 16 + rr + hi * 8;
            float pre = acc[p][rr] + zb[o * 1024 + n0 + nl];
            G[((o >> 1) * 160 + col) * 2 + (o & 1)] = leaky(s5[o] * pre + b5[o]);
        }
    }
    __syncthreads();
    for (int q = 0; q < 2; ++q) {
        int lin = tid + q * 256;
        int o = lin >> 3, nl = lin & 7;
        float m = -3.4e38f;
        for (int k = 0; k < KNB2; ++k)
            m = fmaxf(m, G[((o >> 1) * 160 + nl * KNB2 + k) * 2 + (o & 1)]);
        feat[b * 65536 + o * 1024 + n0 + nl] = m;
    }
}

// ---------------------------------------------------------------- head sigmoid
__global__ __launch_bounds__(256) void k_head(const float* __restrict__ feat,
                                              const float* __restrict__ wh,
                                              float* __restrict__ wout) {
    int t = blockIdx.x * 256 + threadIdx.x;        // B*N
    int b = t >> 10, n = t & 1023;
    const float* fb = feat + b * 65536 + n;
    float s = 0.f;
    for (int o = 0; o < 64; ++o) s += fb[o * 1024] * wh[o];
    wout[t] = 1.f / (1.f + expf(-s));
}

// ---------------------------------------------------------------- jet fit
__device__ float blockReduce(float v, float* red, int tid) {
    red[tid] = v; __syncthreads();
    for (int s = 128; s > 0; s >>= 1) {
        if (tid < s) red[tid] += red[tid + s];
        __syncthreads();
    }
    float r = red[0]; __syncthreads();
    return r;
}

__global__ __launch_bounds__(256) void k_fit(const float* __restrict__ pts,
                                             const float* __restrict__ wv,
                                             float* __restrict__ out) {
    __shared__ float red[256];
    int b = blockIdx.x, tid = threadIdx.x;
    const float* xb = pts + b * 3 * NPTS;
    const float* yb = xb + NPTS;
    const float* zb = xb + 2 * NPTS;
    const float* wb = wv + b * NPTS;

    float cnt = 0.f, sax = 0.f, say = 0.f;
    for (int n = tid; n < NPTS; n += 256) {
        cnt += (wb[n] > 0.001f) ? 1.f : 0.f;
        sax += fabsf(xb[n]); say += fabsf(yb[n]);
    }
    float valid = blockReduce(cnt, red, tid);
    float totx  = blockReduce(sax, red, tid);
    float toty  = blockReduce(say, red, tid);
    bool useW = valid > 18.f;
    float h = (totx / (float)NPTS + toty / (float)NPTS) * 0.5f;
    if (fabsf(h) < 1e-4f) h = 0.1f;

    float xtx[21], xty[6];
    for (int q = 0; q < 21; ++q) xtx[q] = 0.f;
    for (int i = 0; i < 6; ++i) xty[i] = 0.f;
    for (int n = tid; n < NPTS; n += 256) {
        float wn = useW ? wb[n] : 1.f;
        float xs = xb[n] / h, ys = yb[n] / h, zz = zb[n];
        float a[6] = {xs, ys, xs * xs, ys * ys, xs * ys, 1.f};
        int q = 0;
        for (int i = 0; i < 6; ++i) {
            xty[i] += a[i] * wn * zz;
            for (int j = i; j < 6; ++j) xtx[q++] += a[i] * wn * a[j];
        }
    }
    float Xr[21], Yr[6];
    for (int q = 0; q < 21; ++q) Xr[q] = blockReduce(xtx[q], red, tid);
    for (int i = 0; i < 6; ++i)  Yr[i] = blockReduce(xty[i], red, tid);

    if (tid == 0) {
        float M[6][7];
        int q = 0;
        for (int i = 0; i < 6; ++i)
            for (int j = i; j < 6; ++j) { M[i][j] = Xr[q]; M[j][i] = Xr[q]; ++q; }
        for (int i = 0; i < 6; ++i) M[i][6] = Yr[i];
        // Gaussian elimination with partial pivoting
        for (int c = 0; c < 6; ++c) {
            int piv = c; float mx = fabsf(M[c][c]);
            for (int r = c + 1; r < 6; ++r)
                if (fabsf(M[r][c]) > mx) { mx = fabsf(M[r][c]); piv = r; }
            if (piv != c)
                for (int cc = 0; cc < 7; ++cc) { float t = M[c][cc]; M[c][cc] = M[piv][cc]; M[piv][cc] = t; }
            float inv = 1.f / M[c][c];
            for (int r = c + 1; r < 6; ++r) {
                float f = M[r][c] * inv;
                for (int cc = c; cc < 7; ++cc) M[r][cc] -= f * M[c][cc];
            }
        }
        float beta[6];
        for (int i = 5; i >= 0; --i) {
            float s = M[i][6];
            for (int j = i + 1; j < 6; ++j) s -= M[i][j] * beta[j];
            beta[i] = s / M[i][i];
        }
        float h2 = h * h;
        float dinv[6] = {1.f / h, 1.f / h, 1.f / h2, 1.f / h2, 1.f / h2, 1.f};
        for (int i = 0; i < 6; ++i) beta[i] *= dinv[i];
        float nx = -beta[0], ny = -beta[1], nz = 1.f;
        float nn = sqrtf(nx * nx + ny * ny + nz * nz);
        out[b * 3 + 0] = nx / nn; out[b * 3 + 1] = ny / nn; out[b * 3 + 2] = nz / nn;
        for (int i = 0; i < 6; ++i) out[48 + b * 6 + i] = beta[i];
    }
}

// ---------------------------------------------------------------- launch
extern "C" void kernel_launch(void* const* d_in, const int* in_sizes, int n_in,
                              void* d_out, int out_size, void* d_ws, size_t ws_size,
                              hipStream_t stream) {
    (void)n_in; (void)out_size; (void)ws_size;
    int cur = 0;
    const float* points = (const float*)d_in[cur++];
    const float* W0 = (const float*)d_in[cur++];
    const float* bn[6][4];
    auto readBn = [&](int l) {
        if (in_sizes[cur] == 256) {
            const float* p = (const float*)d_in[cur++];
            bn[l][0] = p; bn[l][1] = p + 64; bn[l][2] = p + 128; bn[l][3] = p + 192;
        } else {
            for (int j = 0; j < 4; ++j) bn[l][j] = (const float*)d_in[cur++];
        }
    };
    readBn(0);
    const float* W1 = (const float*)d_in[cur++]; readBn(1);
    const float* W2 = (const float*)d_in[cur++]; readBn(2);
    const float* W3 = (const float*)d_in[cur++]; readBn(3);
    const float* W4 = (const float*)d_in[cur++]; readBn(4);
    const float* W5 = (const float*)d_in[cur++]; readBn(5);
    const float* whead = (const float*)d_in[cur++];

    float* ws = (float*)d_ws;
    float* SB   = ws; ws += 768;
    float* W1ap = ws; ws += 4096;  float* W1dp = ws; ws += 4096;
    float* W3ap = ws; ws += 4096;  float* W3dp = ws; ws += 4096;
    float* W5ap = ws; ws += 4096;  float* W5bp = ws; ws += 4096;
    float* W2p  = ws; ws += 4096;  float* W4p  = ws; ws += 4096;
    float* x0p  = ws; ws += 1048576;
    float* y1   = ws; ws += 1048576;  float* y2   = ws; ws += 1048576;
    float* y3a  = ws; ws += 1048576;  float* y3b  = ws; ws += 1048576;
    float* xk1p = ws; ws += 1048576;  float* z5   = ws; ws += 1048576;
    float* feat = ws; ws += 1048576;
    float* xx   = ws; ws += 16384;
    int* idx = (int*)ws;

    PrepArgs pa;
    for (int l = 0; l < 6; ++l) {
        pa.g[l] = bn[l][0]; pa.b[l] = bn[l][1]; pa.m[l] = bn[l][2]; pa.v[l] = bn[l][3];
    }
    pa.W1 = W1; pa.W2 = W2; pa.W3 = W3; pa.W4 = W4; pa.W5 = W5;
    pa.sb = SB;
    pa.W1ap = W1ap; pa.W1dp = W1dp; pa.W3ap = W3ap; pa.W3dp = W3dp;
    pa.W5ap = W5ap; pa.W5bp = W5bp; pa.W2p = W2p; pa.W4p = W4p;

    float* out = (float*)d_out;

    k_prep<<<1, 64, 0, stream>>>(pa);
    k_conv0<<<4096, 256, 0, stream>>>(points, W0, SB, x0p);
    k_xx<<<64, 256, 0, stream>>>(x0p, xx);
    k_gram_topk<<<dim3(64, 16), 256, 65536, stream>>>(x0p, xx, idx);
    k_gemm64<<<512, 256, 0, stream>>>(W1ap, x0p, y1);
    k_gemm64<<<512, 256, 0, stream>>>(W1dp, x0p, y2);
    k_gemm64<<<512, 256, 0, stream>>>(W3ap, x0p, y3a);
    k_gemm64<<<512, 256, 0, stream>>>(W3dp, x0p, y3b);
    k_fused1<<<dim3(256, 16), 256, 0, stream>>>(y1, y2, idx, W2p, SB, xk1p);
    k_gemm64<<<512, 256, 0, stream>>>(W5bp, xk1p, z5);
    k_fused2<<<dim3(128, 16), 256, 0, stream>>>(y3a, y3b, idx, W4p, W5ap, z5, SB, feat);
    k_head<<<64, 256, 0, stream>>>(feat, whead, out + 144);
    k_fit<<<16, 256, 0, stream>>>(points, out + 144, out);
}